// APPNPStack_62371515072805
// MI455X (gfx1250) — compile-verified
//
#include <hip/hip_runtime.h>
#include <hip/hip_bf16.h>
#include <math.h>

#define NNODES 100000
#define NEDGES 3200000
#define IN_DIM 500
#define HID 64
#define OUT_DIM 40
#define ALPHA 0.1f
#define BN_EPS 1e-5f
#define KSLAB 100

typedef float v2f __attribute__((ext_vector_type(2)));
typedef float v8f __attribute__((ext_vector_type(8)));

// ---------------- degree / norm ----------------

__global__ void init_deg_kernel(float* __restrict__ deg) {
    int i = blockIdx.x * blockDim.x + threadIdx.x;
    if (i < NNODES) deg[i] = 1.0f;  // self-loop contributes 1 to in-degree
}

__global__ void count_deg_kernel(const int* __restrict__ ei, float* __restrict__ deg) {
    int stride = gridDim.x * blockDim.x;
    for (int e = blockIdx.x * blockDim.x + threadIdx.x; e < NEDGES; e += stride) {
        int c = ei[NEDGES + e];           // col = edge_index[1]
        unsafeAtomicAdd(&deg[c], 1.0f);   // integer-valued, exact -> deterministic
    }
}

__global__ void finish_dinv_kernel(float* __restrict__ dinv) {
    int i = blockIdx.x * blockDim.x + threadIdx.x;
    if (i < NNODES) dinv[i] = rsqrtf(dinv[i]);  // deg >= 1 always (self loops)
}

__global__ void norm_kernel(const int* __restrict__ ei, const float* __restrict__ dinv,
                            float* __restrict__ norm) {
    int stride = gridDim.x * blockDim.x;
    for (int e = blockIdx.x * blockDim.x + threadIdx.x; e < NEDGES; e += stride) {
        norm[e] = dinv[ei[e]] * dinv[ei[NEDGES + e]];
    }
}

// ---------------- lin1: h0 = x @ W1 + b1 (f32 WMMA 16x16x4) ----------------
// Per block: 8 waves * 16 rows = 128 rows, all 64 output cols (4 tiles of 16).
// W1 staged in LDS as 100x64 K-slabs, K-pair interleaved so each lane's
// B fragment {W1[k][n], W1[k+1][n]} is one aligned ds_load_b64.

__global__ __launch_bounds__(256) void lin1_wmma_kernel(
        const float* __restrict__ x, const float* __restrict__ W1,
        const float* __restrict__ b1, float* __restrict__ h0) {
    __shared__ float w1s[KSLAB * HID];   // [(k/2)*HID + n]*2 + (k&1)
    const int tid  = threadIdx.x;
    const int wave = tid >> 5;
    const int lane = tid & 31;
    const int hi   = lane >> 4;   // half-wave: K-pair select (0: K0/K1, 1: K2/K3)
    const int lm   = lane & 15;
    const int rowTile = blockIdx.x * 128 + wave * 16;
    // Clamp: always-valid load; out-of-range rows are discarded at the store.
    const int m = (rowTile + lm < NNODES) ? (rowTile + lm) : (NNODES - 1);
    const float2* __restrict__ px = (const float2*)(x + (size_t)m * IN_DIM + hi * 2);

    v8f acc0 = {}, acc1 = {}, acc2 = {}, acc3 = {};

    for (int kb = 0; kb < IN_DIM; kb += KSLAB) {
        __syncthreads();
        for (int idx = tid; idx < KSLAB * HID; idx += 256) {
            int k = idx >> 6, n = idx & 63;
            w1s[(((k >> 1) << 6) + n) * 2 + (k & 1)] = W1[kb * HID + idx];
        }
        __syncthreads();

        for (int kk = 0; kk < KSLAB; kk += 4) {
            // A fragment: row m, K = kb+kk+2*hi .. +1 (single b64 global load)
            float2 av = px[(kb + kk) >> 1];
            v2f a; a.x = av.x; a.y = av.y;
            const int kp = ((kk >> 1) + hi) << 6;   // K-pair row base (in HID units)
            {   v2f b = *(const v2f*)&w1s[(kp + lm) * 2];
                acc0 = __builtin_amdgcn_wmma_f32_16x16x4_f32(false, a, false, b,
                         (short)0, acc0, false, false); }
            {   v2f b = *(const v2f*)&w1s[(kp + 16 + lm) * 2];
                acc1 = __builtin_amdgcn_wmma_f32_16x16x4_f32(false, a, false, b,
                         (short)0, acc1, false, false); }
            {   v2f b = *(const v2f*)&w1s[(kp + 32 + lm) * 2];
                acc2 = __builtin_amdgcn_wmma_f32_16x16x4_f32(false, a, false, b,
                         (short)0, acc2, false, false); }
            {   v2f b = *(const v2f*)&w1s[(kp + 48 + lm) * 2];
                acc3 = __builtin_amdgcn_wmma_f32_16x16x4_f32(false, a, false, b,
                         (short)0, acc3, false, false); }
        }
    }

    // C/D layout: VGPR i, lanes 0-15 -> row rowTile+i, lanes 16-31 -> rowTile+8+i
    const int baseRow = rowTile + hi * 8;
    for (int i = 0; i < 8; ++i) {
        int r = baseRow + i;
        if (r < NNODES) {
            float* hr = h0 + (size_t)r * HID;
            hr[lm]      = acc0[i] + b1[lm];
            hr[16 + lm] = acc1[i] + b1[16 + lm];
            hr[32 + lm] = acc2[i] + b1[32 + lm];
            hr[48 + lm] = acc3[i] + b1[48 + lm];
        }
    }
}

// ---------------- APPNP propagation ----------------
// hnext = ALPHA*h0 + (1-ALPHA)*dinv[i]^2*hcur   (self-loop folded in)

__global__ void prop_init_kernel(const float* __restrict__ h0,
                                 const float* __restrict__ hcur,
                                 const float* __restrict__ dinv,
                                 float* __restrict__ hnext) {
    const int total = NNODES * (HID / 4);
    const int stride = gridDim.x * blockDim.x;
    for (int idx = blockIdx.x * blockDim.x + threadIdx.x; idx < total; idx += stride) {
        int node = idx >> 4;
        float d = dinv[node];
        float w = (1.0f - ALPHA) * d * d;
        float4 v0 = ((const float4*)h0)[idx];
        float4 vc = ((const float4*)hcur)[idx];
        float4 r;
        r.x = ALPHA * v0.x + w * vc.x;
        r.y = ALPHA * v0.y + w * vc.y;
        r.z = ALPHA * v0.z + w * vc.z;
        r.w = ALPHA * v0.w + w * vc.w;
        ((float4*)hnext)[idx] = r;
    }
}

// Batch-coalesced edge scatter: each wave loads 32 edges' (row,col,norm) with
// 3 coalesced loads, then processes 2 edges per step (16 lanes x float4 each),
// broadcasting per-edge values via shuffles. NEDGES % 32 == 0.
__global__ __launch_bounds__(256) void prop_edges_kernel(
        const int* __restrict__ ei, const float* __restrict__ norm,
        const float* __restrict__ hcur, float* __restrict__ hnext) {
    const int lane = threadIdx.x & 31;
    const int hi   = lane >> 4;
    const int lm   = lane & 15;
    const int wave = (blockIdx.x * blockDim.x + threadIdx.x) >> 5;
    const int nwaves = (gridDim.x * blockDim.x) >> 5;

    for (int base = wave * 32; base < NEDGES; base += nwaves * 32) {
        int e = base + lane;
        int r = ei[e];
        int c = ei[NEDGES + e];
        float s = (1.0f - ALPHA) * norm[e];
        #pragma unroll 4
        for (int j = 0; j < 16; ++j) {
            int src   = 2 * j + hi;             // half-wave picks its own edge
            int rj    = __shfl(r, src, 32);
            int cj    = __shfl(c, src, 32);
            float sj  = __shfl(s, src, 32);
            float4 v = ((const float4*)(hcur + (size_t)rj * HID))[lm];
            float* dst = hnext + (size_t)cj * HID + 4 * lm;
            unsafeAtomicAdd(dst + 0, sj * v.x);
            unsafeAtomicAdd(dst + 1, sj * v.y);
            unsafeAtomicAdd(dst + 2, sj * v.z);
            unsafeAtomicAdd(dst + 3, sj * v.w);
        }
    }
}

// ---------------- BN + lin2 + log_softmax ----------------
// One wave per row. Lane l holds BN'd channels 2l,2l+1; shuffle-broadcast over
// 64 channels; lanes 0-31 produce logits 0-31, lanes 0-7 produce 32-39.

__global__ __launch_bounds__(256) void final_kernel(
        const float* __restrict__ h, const float* __restrict__ gamma,
        const float* __restrict__ beta, const float* __restrict__ mean,
        const float* __restrict__ var, const float* __restrict__ W2,
        const float* __restrict__ b2, float* __restrict__ out,
        float* __restrict__ emb) {
    const int lane = threadIdx.x & 31;
    const int wave = (blockIdx.x * blockDim.x + threadIdx.x) >> 5;
    const int nwaves = (gridDim.x * blockDim.x) >> 5;
    const int c0 = 2 * lane, c1 = 2 * lane + 1;
    const float g0 = gamma[c0] * rsqrtf(var[c0] + BN_EPS), m0 = mean[c0], be0 = beta[c0];
    const float g1 = gamma[c1] * rsqrtf(var[c1] + BN_EPS), m1 = mean[c1], be1 = beta[c1];

    for (int i = wave; i < NNODES; i += nwaves) {
        float2 hv = ((const float2*)(h + (size_t)i * HID))[lane];
        float t0 = (hv.x - m0) * g0 + be0;
        float t1 = (hv.y - m1) * g1 + be1;

        float e0 = b2[lane];
        float e1 = (lane < 8) ? b2[32 + lane] : 0.0f;
        for (int c = 0; c < HID; ++c) {
            float tsel = (c & 1) ? t1 : t0;
            float tc = __shfl(tsel, c >> 1, 32);
            e0 += tc * W2[c * OUT_DIM + lane];
            if (lane < 8) e1 += tc * W2[c * OUT_DIM + 32 + lane];
        }
        // log_softmax over the 40 logits {e0: lanes 0-31} u {e1: lanes 0-7}
        float mx = fmaxf(e0, (lane < 8) ? e1 : -INFINITY);
        for (int off = 16; off > 0; off >>= 1) mx = fmaxf(mx, __shfl_xor(mx, off, 32));
        float sm = expf(e0 - mx) + ((lane < 8) ? expf(e1 - mx) : 0.0f);
        for (int off = 16; off > 0; off >>= 1) sm += __shfl_xor(sm, off, 32);
        float lse = logf(sm);

        float* orow = out + (size_t)i * OUT_DIM;
        float* erow = emb + (size_t)i * OUT_DIM;
        orow[lane] = e0 - mx - lse;
        erow[lane] = e0;
        if (lane < 8) {
            orow[32 + lane] = e1 - mx - lse;
            erow[32 + lane] = e1;
        }
    }
}

// ---------------- launch ----------------

extern "C" void kernel_launch(void* const* d_in, const int* in_sizes, int n_in,
                              void* d_out, int out_size, void* d_ws, size_t ws_size,
                              hipStream_t stream) {
    const float* x     = (const float*)d_in[0];
    const int*   ei    = (const int*)d_in[1];
    const float* W1    = (const float*)d_in[2];
    const float* b1    = (const float*)d_in[3];
    const float* gamma = (const float*)d_in[4];
    const float* beta  = (const float*)d_in[5];
    const float* mean  = (const float*)d_in[6];
    const float* var   = (const float*)d_in[7];
    const float* W2    = (const float*)d_in[8];
    const float* b2    = (const float*)d_in[9];
    float* out = (float*)d_out;
    float* emb = out + (size_t)NNODES * OUT_DIM;

    float* ws = (float*)d_ws;
    size_t off = 0;
    float* dinv = ws + off; off += ((size_t)NNODES + 255) & ~(size_t)255;
    float* norm = ws + off; off += ((size_t)NEDGES + 255) & ~(size_t)255;
    float* h0   = ws + off; off += (size_t)NNODES * HID;
    float* ha   = ws + off; off += (size_t)NNODES * HID;
    float* hb   = ws + off;

    init_deg_kernel<<<(NNODES + 255) / 256, 256, 0, stream>>>(dinv);
    count_deg_kernel<<<2048, 256, 0, stream>>>(ei, dinv);
    finish_dinv_kernel<<<(NNODES + 255) / 256, 256, 0, stream>>>(dinv);
    norm_kernel<<<2048, 256, 0, stream>>>(ei, dinv, norm);

    lin1_wmma_kernel<<<(NNODES + 127) / 128, 256, 0, stream>>>(x, W1, b1, h0);

    const float* cur = h0;
    float* bufs[2] = {ha, hb};
    for (int k = 0; k < 10; ++k) {
        float* nxt = bufs[k & 1];
        prop_init_kernel<<<2048, 256, 0, stream>>>(h0, cur, dinv, nxt);
        prop_edges_kernel<<<8192, 256, 0, stream>>>(ei, norm, cur, nxt);
        cur = nxt;
    }

    final_kernel<<<1024, 256, 0, stream>>>(cur, gamma, beta, mean, var, W2, b2, out, emb);
}